// ClfBlock_44641890075009
// MI455X (gfx1250) — compile-verified
//
#include <hip/hip_runtime.h>
#include <math.h>

typedef float v2f __attribute__((ext_vector_type(2)));
typedef float v8f __attribute__((ext_vector_type(8)));

__device__ __forceinline__ float lrelu(float v) { return v > 0.f ? v : 0.01f * v; }

// ---------------------------------------------------------------------------
// K-repack: pair-interleave weights so a WMMA B fragment {k,k+1} x col is one
// contiguous 8-byte unit.  dst[(k>>1)*2C + 2c + (k&1)] = src[k*C + c]
//   Wf_p: 256x256 (C=256)   W1_p: 256x64 (C=64)
// ---------------------------------------------------------------------------
__global__ void __launch_bounds__(256) k_repack(const float* __restrict__ Wf,
                                                const float* __restrict__ W1,
                                                float* __restrict__ Wf_p,
                                                float* __restrict__ W1_p) {
  int i = blockIdx.x * 256 + threadIdx.x;
  if (i < 65536) {
    int k = i >> 8, c = i & 255;
    Wf_p[(k >> 1) * 512 + c * 2 + (k & 1)] = Wf[i];
  } else {
    int j = i - 65536;
    if (j < 16384) {
      int k = j >> 6, c = j & 63;
      W1_p[(k >> 1) * 128 + c * 2 + (k & 1)] = W1[j];
    }
  }
}

// ---------------------------------------------------------------------------
// K1: h = x @ W_feat   (N x 256) @ (256 x 256)
// block = 256 thr (8 waves). Block tile 128 rows x 64 cols.
// B panel staged from pair-interleaved Wf_p in two 32 KB halves; each B
// fragment is a single ds_load_b64 (no register shuffles).
// ---------------------------------------------------------------------------
__global__ void __launch_bounds__(256) k_gemm_feat(const float* __restrict__ x,
                                                   const float* __restrict__ Wf_p,
                                                   float* __restrict__ h, int N) {
  __shared__ float Bs[64 * 128];  // 32 KB: 64 k-pairs x (64 cols x 2)
  const int tid = threadIdx.x;
  const int wave = tid >> 5;
  const int lane = tid & 31;
  const int mlane = lane & 15;
  const int hi8 = (lane >> 4) << 3;   // 0 or 8
  const int koff = (lane >> 4) << 1;  // 0 or 2
  const int kph = lane >> 4;          // k-pair offset: 0 or 1
  const int colBase = blockIdx.y * 64;
  const int row0 = blockIdx.x * 128 + wave * 16;

  int arow = row0 + mlane;
  if (arow >= N) arow = N - 1;  // clamp; stores masked below
  const v2f* __restrict__ ap = (const v2f*)(x + (size_t)arow * 256 + koff);

  v8f acc[4];
#pragma unroll
  for (int t = 0; t < 4; ++t) acc[t] = (v8f){};

  for (int kb = 0; kb < 256; kb += 128) {
    __syncthreads();
    // stage 64 k-pair rows x 128 floats from Wf_p panel: 2048 float4
    const int kbp = kb >> 1;
#pragma unroll
    for (int i = 0; i < 8; ++i) {
      int item = tid + i * 256;   // 0..2047
      int kp = item >> 5;         // 0..63
      int f4 = (item & 31) << 2;  // 0..124
      *(float4*)(Bs + kp * 128 + f4) =
          *(const float4*)(Wf_p + (size_t)(kbp + kp) * 512 + colBase * 2 + f4);
    }
    __syncthreads();
#pragma unroll 4
    for (int k = 0; k < 128; k += 4) {
      v2f a = ap[(kb + k) >> 1];
      const float* bb = Bs + ((k >> 1) + kph) * 128 + mlane * 2;
#pragma unroll
      for (int t = 0; t < 4; ++t) {
        v2f b = *(const v2f*)(bb + t * 32);
        acc[t] = __builtin_amdgcn_wmma_f32_16x16x4_f32(
            false, a, false, b, (short)0, acc[t], false, false);
      }
    }
  }
  // C/D layout: VGPR j, lanes 0-15 -> (M=j, N=lane); lanes 16-31 -> (M=j+8)
#pragma unroll
  for (int t = 0; t < 4; ++t) {
#pragma unroll
    for (int j = 0; j < 8; ++j) {
      int r = row0 + j + hi8;
      if (r < N) h[(size_t)r * 256 + colBase + 16 * t + mlane] = acc[t][j];
    }
  }
}

// ---------------------------------------------------------------------------
// K2: a_s[n,hh] = <h[n,hh,:], phi_s[hh]>, a_d analogous. One thread per (n,hh).
// ---------------------------------------------------------------------------
__global__ void __launch_bounds__(256) k_attn(const float* __restrict__ h,
                                              const float* __restrict__ phi,
                                              float* __restrict__ a_s,
                                              float* __restrict__ a_d, int N) {
  int i = blockIdx.x * 256 + threadIdx.x;
  if (i >= N * 4) return;
  int n = i >> 2, hh = i & 3;
  const float4* hp = (const float4*)(h + (size_t)n * 256 + hh * 64);
  const float4* ps = (const float4*)(phi + hh * 128);
  const float4* pd = (const float4*)(phi + hh * 128 + 64);
  float s = 0.f, d = 0.f;
#pragma unroll
  for (int j = 0; j < 16; ++j) {
    float4 hv = hp[j], sv = ps[j], dv = pd[j];
    s += hv.x * sv.x + hv.y * sv.y + hv.z * sv.z + hv.w * sv.w;
    d += hv.x * dv.x + hv.y * dv.y + hv.z * dv.z + hv.w * dv.w;
  }
  a_s[i] = s;
  a_d[i] = d;
}

// ---------------------------------------------------------------------------
// K0: zero a contiguous float region (agg + group sums + group counts)
// ---------------------------------------------------------------------------
__global__ void __launch_bounds__(256) k_zero(float* __restrict__ p, size_t n) {
  for (size_t i = blockIdx.x * 256ull + threadIdx.x; i < n;
       i += (size_t)gridDim.x * 256ull)
    p[i] = 0.f;
}

// ---------------------------------------------------------------------------
// K3a: per-edge attention weights w[e,hh] = sigmoid(lrelu(a_s[src]+a_d[dst]))
// ---------------------------------------------------------------------------
__global__ void __launch_bounds__(256) k_edge_w(const int* __restrict__ src,
                                                const int* __restrict__ dst,
                                                const float* __restrict__ a_s,
                                                const float* __restrict__ a_d,
                                                float* __restrict__ we, int E) {
  int i = blockIdx.x * 256 + threadIdx.x;
  if (i >= 4 * E) return;
  int e = i >> 2, hh = i & 3;
  float v = a_s[src[e] * 4 + hh] + a_d[dst[e] * 4 + hh];
  v = lrelu(v);
  we[i] = 1.f / (1.f + expf(-v));
}

// ---------------------------------------------------------------------------
// K3b: agg[src] += h[dst] * w   (64 threads x float4 per edge, 4 edges/block)
// L2-resident: h + agg together ~102 MB < 192 MB L2.
// ---------------------------------------------------------------------------
__global__ void __launch_bounds__(256) k_scatter(const int* __restrict__ src,
                                                 const int* __restrict__ dst,
                                                 const float* __restrict__ h,
                                                 const float* __restrict__ we,
                                                 float* __restrict__ agg, int E) {
  int tid = threadIdx.x;
  int e = blockIdx.x * 4 + (tid >> 6);
  if (e >= E) return;
  int sub = tid & 63;
  int s = src[e], d = dst[e];
  float w = we[e * 4 + (sub >> 4)];
  float4 hv = ((const float4*)h)[(size_t)d * 64 + sub];
  float* ag = agg + (size_t)s * 256 + sub * 4;
  atomicAdd(ag + 0, hv.x * w);
  atomicAdd(ag + 1, hv.y * w);
  atomicAdd(ag + 2, hv.z * w);
  atomicAdd(ag + 3, hv.w * w);
}

// ---------------------------------------------------------------------------
// K5: u = lrelu(agg@W1 + b1); z = u@W2 + b2; EPS-softmax -> preds;
//     atomically accumulate group sums / counts.
// B fragments come straight from pair-interleaved W1_p: one global_load_b64
// each (64 KB, cache-resident, shared by all blocks).
// ---------------------------------------------------------------------------
__global__ void __launch_bounds__(256) k_classify(
    const float* __restrict__ agg, const float* __restrict__ W1_p,
    const float* __restrict__ b1, const float* __restrict__ W2,
    const float* __restrict__ b2, const int* __restrict__ batch,
    float* __restrict__ preds, float* __restrict__ gsum,
    float* __restrict__ gcnt, int N) {
  __shared__ float Us[8 * 16 * 64];  // per-wave 16x64 tile, col-major [col][m]
  __shared__ float W2s[640];
  __shared__ float b1s[64];
  __shared__ float b2s[10];
  const int tid = threadIdx.x;
  const int wave = tid >> 5;
  const int lane = tid & 31;
  const int mlane = lane & 15;
  const int hi8 = (lane >> 4) << 3;
  const int koff = (lane >> 4) << 1;
  const int kph = lane >> 4;
  if (tid < 640) W2s[tid] = W2[tid];
  if (tid < 64) b1s[tid] = b1[tid];
  if (tid < 10) b2s[tid] = b2[tid];
  __syncthreads();

  const int row0 = blockIdx.x * 128 + wave * 16;
  int arow = row0 + mlane;
  if (arow >= N) arow = N - 1;
  const v2f* __restrict__ ap = (const v2f*)(agg + (size_t)arow * 256 + koff);

  v8f acc[4];
#pragma unroll
  for (int t = 0; t < 4; ++t) acc[t] = (v8f){};

#pragma unroll 4
  for (int k = 0; k < 256; k += 4) {
    v2f a = ap[k >> 1];
    const float* bb = W1_p + (size_t)((k >> 1) + kph) * 128 + mlane * 2;
#pragma unroll
    for (int t = 0; t < 4; ++t) {
      v2f b = *(const v2f*)(bb + t * 32);
      acc[t] = __builtin_amdgcn_wmma_f32_16x16x4_f32(
          false, a, false, b, (short)0, acc[t], false, false);
    }
  }

  // bias + leaky, stage u col-major so the per-node read below is conflict-free
  float* u = Us + wave * 1024;
#pragma unroll
  for (int t = 0; t < 4; ++t) {
#pragma unroll
    for (int j = 0; j < 8; ++j) {
      int col = 16 * t + mlane;
      int m = j + hi8;
      u[col * 16 + m] = lrelu(acc[t][j] + b1s[col]);
    }
  }
  __syncthreads();

  if (lane < 16) {
    int node = row0 + lane;
    if (node < N) {
      float zz[10];
#pragma unroll
      for (int c = 0; c < 10; ++c) zz[c] = b2s[c];
      for (int d = 0; d < 64; ++d) {
        float uv = u[d * 16 + lane];
#pragma unroll
        for (int c = 0; c < 10; ++c) zz[c] += uv * W2s[d * 10 + c];
      }
      float mx = zz[0];
#pragma unroll
      for (int c = 1; c < 10; ++c) mx = fmaxf(mx, zz[c]);
      float s = 0.f;
#pragma unroll
      for (int c = 0; c < 10; ++c) {
        zz[c] = expf(zz[c] - mx) + 1e-4f;
        s += zz[c];
      }
      float inv = 1.f / s;
      int b = batch[node];
#pragma unroll
      for (int c = 0; c < 10; ++c) {
        float p = zz[c] * inv;
        preds[(size_t)node * 10 + c] = p;
        atomicAdd(&gsum[b * 10 + c], p);
      }
      atomicAdd(&gcnt[b], 1.0f);
    }
  }
}

// ---------------------------------------------------------------------------
// K6: log(group mean)
// ---------------------------------------------------------------------------
__global__ void __launch_bounds__(256) k_final(const float* __restrict__ gsum,
                                               const float* __restrict__ gcnt,
                                               float* __restrict__ out) {
  int i = blockIdx.x * 256 + threadIdx.x;
  if (i >= 1000) return;
  out[i] = logf(gsum[i] / gcnt[i / 10]);
}

// ---------------------------------------------------------------------------
extern "C" void kernel_launch(void* const* d_in, const int* in_sizes, int n_in,
                              void* d_out, int out_size, void* d_ws,
                              size_t ws_size, hipStream_t stream) {
  const float* x = (const float*)d_in[0];
  const int* midx = (const int*)d_in[1];
  const int* batch = (const int*)d_in[2];
  const float* W_feat = (const float*)d_in[3];
  const float* phi = (const float*)d_in[4];
  const float* W1 = (const float*)d_in[5];
  const float* b1 = (const float*)d_in[6];
  const float* W2 = (const float*)d_in[7];
  const float* b2 = (const float*)d_in[8];
  float* out = (float*)d_out;

  const int N = in_sizes[0] / 256;
  const int E = in_sizes[1] / 2;
  const int* src = midx;
  const int* dst = midx + E;

  float* ws = (float*)d_ws;
  size_t off = 0;
  float* h = ws + off;     off += (size_t)N * 256;
  float* a_s = ws + off;   off += (size_t)N * 4;
  float* a_d = ws + off;   off += (size_t)N * 4;
  float* we = ws + off;    off += (size_t)E * 4;
  float* agg = ws + off;   off += (size_t)N * 256;
  float* gsum = ws + off;  off += 1000;
  float* gcnt = ws + off;  off += 100;
  float* Wf_p = ws + off;  off += 65536;
  float* W1_p = ws + off;  off += 16384;
  (void)off; (void)ws_size; (void)n_in; (void)out_size;

  const int rb = (N + 127) / 128;

  k_repack<<<(65536 + 16384) / 256, 256, 0, stream>>>(W_feat, W1, Wf_p, W1_p);
  k_gemm_feat<<<dim3(rb, 4), 256, 0, stream>>>(x, Wf_p, h, N);
  k_attn<<<(N * 4 + 255) / 256, 256, 0, stream>>>(h, phi, a_s, a_d, N);
  k_zero<<<2048, 256, 0, stream>>>(agg, (size_t)N * 256 + 1100);
  k_edge_w<<<((size_t)4 * E + 255) / 256, 256, 0, stream>>>(src, dst, a_s, a_d,
                                                            we, E);
  k_scatter<<<(E + 3) / 4, 256, 0, stream>>>(src, dst, h, we, agg, E);
  k_classify<<<rb, 256, 0, stream>>>(agg, W1_p, b1, W2, b2, batch, out + 1000,
                                     gsum, gcnt, N);
  k_final<<<4, 256, 0, stream>>>(gsum, gcnt, out);
}